// LSM_30176440221725
// MI455X (gfx1250) — compile-verified
//
#include <hip/hip_runtime.h>

typedef __attribute__((ext_vector_type(2))) float v2f;
typedef __attribute__((ext_vector_type(8))) float v8f;

#define EPS_F 1e-6f
#define NEG_BIG -1e30f
#define LOG2E_F 1.4426950408889634f

constexpr int SI = 3000;
constexpr int SJ = 3000;
constexpr int SP = 3008;          // padded to multiple of 16
constexpr int TILES = SP / 16;    // 188
constexpr int TCOLP = TILES / 2;  // 94 column-pairs (16x32 strips)
constexpr int NSTRIPS = TILES * TCOLP;

constexpr int GRID_A = 512;       // pair-kernel blocks
constexpr int GRID_L = 512;       // links-kernel blocks

// workspace layout (in floats)
constexpr int OFF_ZI = 0;                 // [SP*8]  gathered zi + eps
constexpr int OFF_ZJ = SP * 8;            // [SP*8]  gathered zj
constexpr int OFF_NI = 2 * SP * 8;        // [SP]    ||zi'||^2
constexpr int OFF_NJ = OFF_NI + SP;       // [SP]    ||zj||^2
constexpr int OFF_BI = OFF_NJ + SP;       // [SP]    beta  (pad = -1e30)
constexpr int OFF_GJ = OFF_BI + SP;       // [SP]    gamma (pad = -1e30)
constexpr int OFF_PA = OFF_GJ + SP;       // [GRID_A] partial exp-sums
constexpr int OFF_PL = OFF_PA + GRID_A;   // [GRID_L] partial link-sums

// ---------------------------------------------------------------------------
// Kernel 1: gather sampled rows, add EPS on the i-side, compute norms, pad.
// ---------------------------------------------------------------------------
__global__ void __launch_bounds__(256)
lsm_prep(const float* __restrict__ zi, const float* __restrict__ zj,
         const float* __restrict__ beta, const float* __restrict__ gamma,
         const int* __restrict__ si, const int* __restrict__ sj,
         float* __restrict__ ws) {
  int t = blockIdx.x * blockDim.x + threadIdx.x;
  if (t >= 2 * SP) return;
  bool isJ = (t >= SP);
  int s = isJ ? (t - SP) : t;

  float* Z = ws + (isJ ? OFF_ZJ : OFF_ZI);
  float* N = ws + (isJ ? OFF_NJ : OFF_NI);
  float* B = ws + (isJ ? OFF_GJ : OFF_BI);
  int lim = isJ ? SJ : SI;

  if (s < lim) {
    int idx = isJ ? sj[s] : si[s];
    const float* src = (isJ ? zj : zi) + (size_t)idx * 8;
    float n = 0.0f;
    float r[8];
#pragma unroll
    for (int d = 0; d < 8; ++d) {
      float v = src[d];
      if (!isJ) v += EPS_F;   // fold +EPS into the i-side rows
      r[d] = v;
      n += v * v;
    }
#pragma unroll
    for (int d = 0; d < 8; ++d) Z[s * 8 + d] = r[d];
    N[s] = n;
    B[s] = isJ ? gamma[idx] : beta[idx];
  } else {
    // padding: zero rows, beta/gamma = -1e30 so exp(Lambda) == 0 exactly
#pragma unroll
    for (int d = 0; d < 8; ++d) Z[s * 8 + d] = 0.0f;
    N[s] = 0.0f;
    B[s] = NEG_BIG;
  }
}

// ---------------------------------------------------------------------------
// Kernel 2: pairwise block term via V_WMMA_F32_16X16X4_F32.
// One wave per 16x32 strip (two 16x16 tiles): A fragments + row-side scalars
// are loaded once and amortized over 4 WMMAs. K=8 => two K=4 WMMAs per tile.
// d2 = ni + nj - 2*G, Lambda = bi + gj - sqrt(d2), accumulate exp(Lambda)
// using hardware v_sqrt_f32 / v_exp_f32 (inputs are range-safe).
// ---------------------------------------------------------------------------
__global__ void __launch_bounds__(256)
lsm_pair(const float* __restrict__ ws, float* __restrict__ partA) {
  const float* ZiP = ws + OFF_ZI;
  const float* ZjP = ws + OFF_ZJ;
  const float* ni  = ws + OFF_NI;
  const float* nj  = ws + OFF_NJ;
  const float* bi  = ws + OFF_BI;
  const float* gj  = ws + OFF_GJ;

  const int lane = threadIdx.x & 31;
  const int wave = threadIdx.x >> 5;
  const int wavesPerBlock = blockDim.x >> 5;
  const int gwave  = blockIdx.x * wavesPerBlock + wave;
  const int nwaves = gridDim.x * wavesPerBlock;

  const int half = lane >> 4;      // 0: lanes 0-15, 1: lanes 16-31
  const int l16  = lane & 15;
  const int koff = half * 2;       // A/B fragment K-offset per ISA layout

  float lsum = 0.0f;

  for (int t = gwave; t < NSTRIPS; t += nwaves) {
    const int tm  = t / TCOLP;
    const int tnp = t - tm * TCOLP;
    const int tn0 = 2 * tnp;
    const int tn1 = tn0 + 1;
    const int arow  = tm * 16 + l16;
    const int bcol0 = tn0 * 16 + l16;
    const int bcol1 = tn1 * 16 + l16;

    // A fragment: row-major zi' rows; VGPR0/1 hold K={koff,koff+1}
    v2f aLo = *(const v2f*)(ZiP + arow * 8 + koff);      // K 0..3
    v2f aHi = *(const v2f*)(ZiP + arow * 8 + 4 + koff);  // K 4..7
    // B fragments for the two column tiles
    v2f b0Lo = *(const v2f*)(ZjP + bcol0 * 8 + koff);
    v2f b0Hi = *(const v2f*)(ZjP + bcol0 * 8 + 4 + koff);
    v2f b1Lo = *(const v2f*)(ZjP + bcol1 * 8 + koff);
    v2f b1Hi = *(const v2f*)(ZjP + bcol1 * 8 + 4 + koff);

    v8f acc0 = {};
    v8f acc1 = {};
    acc0 = __builtin_amdgcn_wmma_f32_16x16x4_f32(false, aLo, false, b0Lo,
                                                 (short)0, acc0, false, false);
    acc1 = __builtin_amdgcn_wmma_f32_16x16x4_f32(false, aLo, false, b1Lo,
                                                 (short)0, acc1, false, false);
    acc0 = __builtin_amdgcn_wmma_f32_16x16x4_f32(false, aHi, false, b0Hi,
                                                 (short)0, acc0, false, false);
    acc1 = __builtin_amdgcn_wmma_f32_16x16x4_f32(false, aHi, false, b1Hi,
                                                 (short)0, acc1, false, false);

    // Row-side scalars, shared by both column tiles.
    // C/D layout: VGPR r, this lane -> M = tm*16 + r + 8*half, N = tile_n*16 + l16
    const int mbase = tm * 16 + 8 * half;
    const float4* nip = (const float4*)(ni + mbase);
    const float4* bip = (const float4*)(bi + mbase);
    float4 n0 = nip[0], n1 = nip[1];
    float4 b0 = bip[0], b1 = bip[1];
    float nn[8] = {n0.x, n0.y, n0.z, n0.w, n1.x, n1.y, n1.z, n1.w};
    float bb[8] = {b0.x, b0.y, b0.z, b0.w, b1.x, b1.y, b1.z, b1.w};

    const float njv0 = nj[bcol0], gjv0 = gj[bcol0];
    const float njv1 = nj[bcol1], gjv1 = gj[bcol1];

#pragma unroll
    for (int r = 0; r < 8; ++r) {
      float d20 = fmaxf(nn[r] + njv0 - 2.0f * acc0[r], 0.0f);
      float d21 = fmaxf(nn[r] + njv1 - 2.0f * acc1[r], 0.0f);
      float lam0 = bb[r] + gjv0 - __builtin_amdgcn_sqrtf(d20);
      float lam1 = bb[r] + gjv1 - __builtin_amdgcn_sqrtf(d21);
      // exp(lam) = 2^(lam * log2(e)); pad rows give 2^(-inf-ish) == 0
      lsum += __builtin_amdgcn_exp2f(lam0 * LOG2E_F);
      lsum += __builtin_amdgcn_exp2f(lam1 * LOG2E_F);
    }
  }

  // deterministic block tree reduction
  __shared__ float red[256];
  red[threadIdx.x] = lsum;
  __syncthreads();
  for (int s = 128; s > 0; s >>= 1) {
    if (threadIdx.x < s) red[threadIdx.x] += red[threadIdx.x + s];
    __syncthreads();
  }
  if (threadIdx.x == 0) partA[blockIdx.x] = red[0];
}

// ---------------------------------------------------------------------------
// Kernel 3: link (edge-list) term. L2-resident random gathers.
// ---------------------------------------------------------------------------
__global__ void __launch_bounds__(256)
lsm_links(const float* __restrict__ zi, const float* __restrict__ zj,
          const float* __restrict__ beta, const float* __restrict__ gamma,
          const int* __restrict__ ei, const int* __restrict__ ej,
          int nE, float* __restrict__ partL) {
  int tid = blockIdx.x * blockDim.x + threadIdx.x;
  int stride = gridDim.x * blockDim.x;
  float lsum = 0.0f;

  for (int e = tid; e < nE; e += stride) {
    int i = ei[e];
    int j = ej[e];
    const float4* a = (const float4*)(zi + (size_t)i * 8);
    const float4* b = (const float4*)(zj + (size_t)j * 8);
    float4 a0 = a[0], a1 = a[1];
    float4 b0 = b[0], b1 = b[1];
    float d2 = 0.0f, dx;
    dx = a0.x - b0.x + EPS_F; d2 += dx * dx;
    dx = a0.y - b0.y + EPS_F; d2 += dx * dx;
    dx = a0.z - b0.z + EPS_F; d2 += dx * dx;
    dx = a0.w - b0.w + EPS_F; d2 += dx * dx;
    dx = a1.x - b1.x + EPS_F; d2 += dx * dx;
    dx = a1.y - b1.y + EPS_F; d2 += dx * dx;
    dx = a1.z - b1.z + EPS_F; d2 += dx * dx;
    dx = a1.w - b1.w + EPS_F; d2 += dx * dx;
    lsum += beta[i] + gamma[j] - __builtin_amdgcn_sqrtf(d2);
  }

  __shared__ float red[256];
  red[threadIdx.x] = lsum;
  __syncthreads();
  for (int s = 128; s > 0; s >>= 1) {
    if (threadIdx.x < s) red[threadIdx.x] += red[threadIdx.x + s];
    __syncthreads();
  }
  if (threadIdx.x == 0) partL[blockIdx.x] = red[0];
}

// ---------------------------------------------------------------------------
// Kernel 4: final deterministic reduction. LL = sum(links) - sum(exp block).
// ---------------------------------------------------------------------------
__global__ void __launch_bounds__(256)
lsm_finish(const float* __restrict__ ws, float* __restrict__ out) {
  __shared__ float sA[256];
  __shared__ float sL[256];
  int t = threadIdx.x;
  float a = 0.0f, l = 0.0f;
  for (int k = t; k < GRID_A; k += 256) a += ws[OFF_PA + k];
  for (int k = t; k < GRID_L; k += 256) l += ws[OFF_PL + k];
  sA[t] = a; sL[t] = l;
  __syncthreads();
  for (int s = 128; s > 0; s >>= 1) {
    if (t < s) { sA[t] += sA[t + s]; sL[t] += sL[t + s]; }
    __syncthreads();
  }
  if (t == 0) out[0] = sL[0] - sA[0];
}

// ---------------------------------------------------------------------------
extern "C" void kernel_launch(void* const* d_in, const int* in_sizes, int n_in,
                              void* d_out, int out_size, void* d_ws, size_t ws_size,
                              hipStream_t stream) {
  const float* zi    = (const float*)d_in[0];  // [N_I, 8]
  const float* zj    = (const float*)d_in[1];  // [N_J, 8]
  const float* beta  = (const float*)d_in[2];  // [N_I]
  const float* gamma = (const float*)d_in[3];  // [N_J]
  const int*   si    = (const int*)d_in[4];    // [S_I]
  const int*   sj    = (const int*)d_in[5];    // [S_J]
  const int*   ei    = (const int*)d_in[6];    // [E]
  const int*   ej    = (const int*)d_in[7];    // [E]
  const int nE = in_sizes[6];

  float* ws  = (float*)d_ws;
  float* out = (float*)d_out;

  lsm_prep<<<(2 * SP + 255) / 256, 256, 0, stream>>>(zi, zj, beta, gamma, si, sj, ws);
  lsm_pair<<<GRID_A, 256, 0, stream>>>(ws, ws + OFF_PA);
  lsm_links<<<GRID_L, 256, 0, stream>>>(zi, zj, beta, gamma, ei, ej, nE, ws + OFF_PL);
  lsm_finish<<<1, 256, 0, stream>>>(ws, out);
}